// HumanAwareRouter_8512625180954
// MI455X (gfx1250) — compile-verified
//
#include <hip/hip_runtime.h>
#include <hip/hip_bf16.h>
#include <cstdint>

#define BATCH 32
#define IMH 512
#define IMW 512
#define HWSZ (IMH * IMW)
#define TILE 32
#define TW 34              // TILE + 2*halo (columns tx0-1 .. tx0+32)
#define TSZ (TW * TW)      // 1156 staged elements
#define LSTRIDE 40         // padded LDS row stride: interior chunk 16B-aligned
#define LN (TW * LSTRIDE)  // 1360 floats per LDS plane
#define KROUTES 256
#define STRIDE_F 4.0f
#define MIN_SIZE 32.0f
#define MAX_SIZE 512.0f

__device__ __forceinline__ float sigmoidf(float x) {
    return 1.0f / (1.0f + __expf(-x));
}

__device__ __forceinline__ void async_wait0() {
#if __has_builtin(__builtin_amdgcn_s_wait_asynccnt)
    __builtin_amdgcn_s_wait_asynccnt(0);
#else
    asm volatile("s_wait_asynccnt 0" ::: "memory");
#endif
}

// CDNA5 async global->LDS (ASYNCcnt-tracked). VDST = per-lane LDS byte
// address, VADDR = 64-bit global address, no SADDR.
__device__ __forceinline__ void async_copy_b32(unsigned lds_byte_addr,
                                               const float* gptr) {
    unsigned long long ga = (unsigned long long)(uintptr_t)gptr;
    asm volatile("global_load_async_to_lds_b32 %0, %1, off"
                 :: "v"(lds_byte_addr), "v"(ga)
                 : "memory");
}

__device__ __forceinline__ void async_copy_b128(unsigned lds_byte_addr,
                                                const float* gptr) {
    unsigned long long ga = (unsigned long long)(uintptr_t)gptr;
    asm volatile("global_load_async_to_lds_b128 %0, %1, off"
                 :: "v"(lds_byte_addr), "v"(ga)
                 : "memory");
}

__global__ void route_nms_init(unsigned* counters) {
    if (threadIdx.x < BATCH) counters[threadIdx.x] = 0u;
}

// Pass 1: fused score + 3x3 local-max NMS, candidate emission.
// LDS plane layout: element (ly, j), j in [0,34) covering columns tx0-1+j,
// lives at ly*LSTRIDE + 3 + j  => interior 32-col chunk starts at +4 (16B
// aligned), halos at +3 and +36.
__global__ __launch_bounds__(256) void route_nms_pass1(
        const float* __restrict__ route,
        const float* __restrict__ unc,
        unsigned* __restrict__ counters,
        uint2* __restrict__ cand,
        unsigned cap) {
    __shared__ __align__(16) float s_route[LN];
    __shared__ __align__(16) float s_unc[LN];
    __shared__ __align__(16) float s_score[LN];

    const int tid = threadIdx.x;
    const int tx0 = blockIdx.x * TILE;
    const int ty0 = blockIdx.y * TILE;
    const int b   = blockIdx.z;
    const size_t base = (size_t)b * HWSZ;

    const bool interior = (blockIdx.x > 0) && (blockIdx.x < IMW / TILE - 1) &&
                          (blockIdx.y > 0) && (blockIdx.y < IMH / TILE - 1);

    if (interior) {
        // Fast path: 16B-aligned b128 async staging of the 34x32 interior
        // (8 chunks/row), b32 for the two halo columns. No clamping needed.
        for (int i = tid; i < TW * 8; i += 256) {     // 272 (row, chunk) pairs
            int ly = i >> 3, c = i & 7;
            size_t g = base + (size_t)(ty0 - 1 + ly) * IMW + (tx0 + 4 * c);
            unsigned lp = (unsigned)(ly * LSTRIDE + 4 + 4 * c) * 4u;
            async_copy_b128((unsigned)(uintptr_t)s_route + lp, route + g);
            async_copy_b128((unsigned)(uintptr_t)s_unc + lp,   unc + g);
        }
        if (tid < TW * 2) {                            // 68 halo dwords
            int ly = tid >> 1;
            int right = tid & 1;
            int gx = right ? (tx0 + TILE) : (tx0 - 1);
            size_t g = base + (size_t)(ty0 - 1 + ly) * IMW + gx;
            unsigned lp = (unsigned)(ly * LSTRIDE + (right ? 36 : 3)) * 4u;
            async_copy_b32((unsigned)(uintptr_t)s_route + lp, route + g);
            async_copy_b32((unsigned)(uintptr_t)s_unc + lp,   unc + g);
        }
    } else {
        // Border tiles: per-element b32 with image-border clamping (clamping
        // only duplicates in-window values => identical to -inf pooling pad).
        for (int i = tid; i < TSZ; i += 256) {
            int ly = i / TW, j = i - ly * TW;
            int gy = min(max(ty0 - 1 + ly, 0), IMH - 1);
            int gx = min(max(tx0 - 1 + j, 0), IMW - 1);
            size_t g = base + (size_t)gy * IMW + gx;
            unsigned lp = (unsigned)(ly * LSTRIDE + 3 + j) * 4u;
            async_copy_b32((unsigned)(uintptr_t)s_route + lp, route + g);
            async_copy_b32((unsigned)(uintptr_t)s_unc + lp,   unc + g);
        }
    }
    async_wait0();
    __syncthreads();

    // score = sigmoid(route)^2 * (1 - 0.35*sigmoid(unc)); computed once so
    // the center/pooled equality below is bit-exact.
    for (int i = tid; i < TSZ; i += 256) {
        int ly = i / TW, j = i - ly * TW;
        int p = ly * LSTRIDE + 3 + j;
        float s = sigmoidf(s_route[p]);
        float u = sigmoidf(s_unc[p]);
        s_score[p] = s * s * (1.0f - 0.35f * u);
    }
    __syncthreads();

    for (int i = tid; i < TILE * TILE; i += 256) {
        int iy = i >> 5, ix = i & 31;
        int c = (iy + 1) * LSTRIDE + 4 + ix;
        float v = s_score[c];
        float m = v;
        m = fmaxf(m, s_score[c - LSTRIDE - 1]);
        m = fmaxf(m, s_score[c - LSTRIDE]);
        m = fmaxf(m, s_score[c - LSTRIDE + 1]);
        m = fmaxf(m, s_score[c - 1]);
        m = fmaxf(m, s_score[c + 1]);
        m = fmaxf(m, s_score[c + LSTRIDE - 1]);
        m = fmaxf(m, s_score[c + LSTRIDE]);
        m = fmaxf(m, s_score[c + LSTRIDE + 1]);
        if (v >= m && v > 0.0f) {   // v == pooled local max, top_k filter > 0
            unsigned flat = (unsigned)((ty0 + iy) * IMW + (tx0 + ix));
            unsigned p = atomicAdd(&counters[b], 1u);
            if (p < cap) cand[(size_t)b * cap + p] =
                make_uint2(__float_as_uint(v), flat);
        }
    }
}

// Pass 2: exact per-image top-256 (radix select + bitonic sort), ROI decode.
__global__ __launch_bounds__(256) void route_nms_pass2(
        const unsigned* __restrict__ counters,
        const uint2* __restrict__ cand,
        const float* __restrict__ scale_logits,
        const float* __restrict__ unc_logits,
        const int* __restrict__ p_img_h,
        const int* __restrict__ p_img_w,
        float* __restrict__ out,
        unsigned cap) {
    __shared__ unsigned long long buf[1024];
    __shared__ unsigned hist[256];
    __shared__ unsigned sh_sel, sh_K, sh_nsel;

    const int b = blockIdx.x;
    const int tid = threadIdx.x;
    const unsigned N = min(counters[b], cap);
    const uint2* cb = cand + (size_t)b * cap;

    // ---- radix select: T = 256th-largest score bits (positive floats are
    // order-preserving as u32) ----
    unsigned T = 0;
    if (N > KROUTES) {
        unsigned prefix = 0, pmask = 0, K = KROUTES;
        for (int shift = 24; shift >= 0; shift -= 8) {
            hist[tid] = 0u;
            __syncthreads();
            for (unsigned i = tid; i < N; i += 256) {
                unsigned k = cb[i].x;
                if ((k & pmask) == prefix)
                    atomicAdd(&hist[(k >> shift) & 255u], 1u);
            }
            __syncthreads();
            if (tid == 0) {
                unsigned cum = 0, sel = 0, Kn = K;
                for (int bb = 255; bb >= 0; --bb) {
                    unsigned c = hist[bb];
                    if (cum + c >= K) { sel = (unsigned)bb; Kn = K - cum; break; }
                    cum += c;
                }
                sh_sel = sel; sh_K = Kn;
            }
            __syncthreads();
            prefix |= (sh_sel << shift);
            pmask  |= (0xFFu << shift);
            K = sh_K;
            __syncthreads();
        }
        T = prefix;
    }

    // ---- collect candidates >= T (all if N <= 256) ----
    for (int i = tid; i < 1024; i += 256) buf[i] = 0ull;
    if (tid == 0) sh_nsel = 0u;
    __syncthreads();
    for (unsigned i = tid; i < N; i += 256) {
        uint2 c = cb[i];
        bool take = (N <= KROUTES) || (c.x >= T);
        if (take) {
            unsigned p = atomicAdd(&sh_nsel, 1u);
            if (p < 1024u)
                buf[p] = ((unsigned long long)c.x << 32) | (unsigned)(~c.y);
        }
    }
    __syncthreads();

    // ---- bitonic sort, descending on (score, ~index): score desc, idx asc ----
    for (int k = 2; k <= 1024; k <<= 1) {
        for (int j = k >> 1; j > 0; j >>= 1) {
            for (int i = tid; i < 1024; i += 256) {
                int ixj = i ^ j;
                if (ixj > i) {
                    unsigned long long a = buf[i], c = buf[ixj];
                    bool sw = ((i & k) == 0) ? (a < c) : (a > c);
                    if (sw) { buf[i] = c; buf[ixj] = a; }
                }
            }
            __syncthreads();
        }
    }

    // ---- ROI decode + outputs: [B,256,5] rois | [B,256] scores | [B,256] valid ----
    const float img_h = (float)(*p_img_h);
    const float img_w = (float)(*p_img_w);
    float* rois   = out;
    float* scores = out + (size_t)BATCH * KROUTES * 5;
    float* validf = out + (size_t)BATCH * KROUTES * 6;

    if (tid < KROUTES) {
        int s = tid;
        unsigned long long key = buf[s];
        float sc = __uint_as_float((unsigned)(key >> 32));
        unsigned flat = ~(unsigned)key;
        bool valid = (sc > 0.0f) && (flat < (unsigned)HWSZ);
        float ro0 = 0.f, ro1 = 0.f, ro2 = 0.f, ro3 = 0.f, ro4 = 0.f;
        float sv = 0.f, vv = 0.f;
        if (valid) {
            int y = (int)(flat / IMW), x = (int)(flat % IMW);
            float cx = ((float)x + 0.5f) * STRIDE_F;
            float cy = ((float)y + 0.5f) * STRIDE_F;
            size_t g = (size_t)b * HWSZ + flat;
            float sg = sigmoidf(scale_logits[g]);
            float uu = sigmoidf(unc_logits[g]);
            float side = (MIN_SIZE + sg * (MAX_SIZE - MIN_SIZE)) * (1.0f + 0.25f * uu);
            float half = 0.5f * side;
            ro0 = (float)b;
            ro1 = fminf(fmaxf(cx - half, 0.0f), img_w - 1.0f);
            ro2 = fminf(fmaxf(cy - half, 0.0f), img_h - 1.0f);
            ro3 = fminf(fmaxf(cx + half, 1.0f), img_w);
            ro4 = fminf(fmaxf(cy + half, 1.0f), img_h);
            sv = sc; vv = 1.0f;
        }
        size_t rb = ((size_t)b * KROUTES + s) * 5;
        rois[rb + 0] = ro0; rois[rb + 1] = ro1; rois[rb + 2] = ro2;
        rois[rb + 3] = ro3; rois[rb + 4] = ro4;
        scores[(size_t)b * KROUTES + s] = sv;
        validf[(size_t)b * KROUTES + s] = vv;
    }
}

extern "C" void kernel_launch(void* const* d_in, const int* in_sizes, int n_in,
                              void* d_out, int out_size, void* d_ws, size_t ws_size,
                              hipStream_t stream) {
    const float* route = (const float*)d_in[0];
    const float* scale = (const float*)d_in[1];
    const float* unc   = (const float*)d_in[2];
    const int* p_img_h = (const int*)d_in[3];
    const int* p_img_w = (const int*)d_in[4];

    // ws layout: [0..255] counters (B u32, padded) | candidate uint2 array
    unsigned* counters = (unsigned*)d_ws;
    uint2* cand = (uint2*)((char*)d_ws + 256);

    // Per-image candidate capacity derived from ws_size (expected maxima
    // ~H*W/9 ~= 29k/image; prefer 64k headroom when workspace allows).
    size_t cap_sz = (ws_size > 4096) ? (ws_size - 256) / ((size_t)BATCH * sizeof(uint2))
                                     : 1;
    unsigned cap = (unsigned)(cap_sz < 65536 ? cap_sz : 65536);
    if (cap == 0) cap = 1;

    route_nms_init<<<1, 32, 0, stream>>>(counters);

    dim3 g1(IMW / TILE, IMH / TILE, BATCH);   // 16 x 16 x 32 blocks
    route_nms_pass1<<<g1, 256, 0, stream>>>(route, unc, counters, cand, cap);

    route_nms_pass2<<<BATCH, 256, 0, stream>>>(counters, cand, scale, unc,
                                               p_img_h, p_img_w,
                                               (float*)d_out, cap);
}